// DMLoss_21723944583646
// MI455X (gfx1250) — compile-verified
//
#include <hip/hip_runtime.h>

typedef __attribute__((ext_vector_type(2))) float v2f;
typedef __attribute__((ext_vector_type(8))) float v8f;

#define BB 1024
#define NN 128
#define MM 128
#define TT 10
#define GG (MM * TT)   // 1280 interpolated gt points

// Branchless (val,idx) argmin combine: pure cndmask, no exec branches.
__device__ __forceinline__ void argminStep(float& v, int& i, float ov, int oi) {
    bool better = (ov < v) | ((ov == v) & (oi < i));
    v = better ? ov : v;
    i = better ? oi : i;
}

// One workgroup (256 threads = 8 waves) per batch.
// Wave w owns row-tile w (16 rows) in both distance phases.
// B operands live in LDS pre-swizzled into the WMMA 16x16x4 lane layout:
// flat float2 array, [col] = (x,y) for K=0,1 lanes (0-15),
// [STRIDE+col] = (|v|^2, 0) for K=2,3 lanes (16-31). Inner loop is one
// branchless ds_load_b64 per tile, software-pipelined one tile ahead.
__global__ __launch_bounds__(256) void dmloss_main_kernel(
    const float* __restrict__ iniPred,   // [B,N,2]
    const float* __restrict__ predPoly,  // [B,N,2]
    const float* __restrict__ gtPoly,    // [B,M,2]
    const float* __restrict__ kpMask,    // [B,M]
    float* __restrict__ ws)              // [3*B] partials
{
    __shared__ float2 sBint[2 * GG];     // interp-gt B operand, pre-swizzled
    __shared__ float2 sBpred[2 * NN];    // ini_pred  B operand, pre-swizzled
    __shared__ float2 sGt[MM];           // raw gt coords
    __shared__ float2 sPredPoly[NN];     // pred_polys_ coords (gather target)
    __shared__ float  c1[NN];            // per-row L1 (pred->gt)
    __shared__ float  c2[MM];            // per-row masked L1 (gt->pred)
    __shared__ float  c3[MM];            // per-row mask*2

    const int b    = blockIdx.x;
    const int tid  = threadIdx.x;
    const int lane = tid & 31;
    const int wave = tid >> 5;
    const int half = lane >> 4;   // 0: K=0,1 lanes ; 1: K=2,3 lanes
    const int l    = lane & 15;

    // ---- stage inputs into LDS ----
    const float* gtb = gtPoly   + b * MM * 2;
    const float* ipb = iniPred  + b * NN * 2;
    const float* ppb = predPoly + b * NN * 2;
    for (int i = tid; i < MM; i += 256)
        sGt[i] = make_float2(gtb[2*i], gtb[2*i+1]);
    for (int i = tid; i < NN; i += 256) {
        float px = ipb[2*i], py = ipb[2*i+1];
        sBpred[i]      = make_float2(px, py);                // K=0,1 half
        sBpred[NN + i] = make_float2(px*px + py*py, 0.f);    // K=2,3 half
        sPredPoly[i]   = make_float2(ppb[2*i], ppb[2*i+1]);
    }
    __syncthreads();

    // ---- interpolation: interp[m*10+t] = gt[m]*(t/10) + gt[m-1]*(1-t/10) ----
    for (int g = tid; g < GG; g += 256) {
        int m = g / TT;
        int t = g - m * TT;
        float w = (float)t * (1.0f / (float)TT);
        float2 cur = sGt[m];
        float2 prv = sGt[(m + MM - 1) & (MM - 1)];
        float x = cur.x * w + prv.x * (1.f - w);
        float y = cur.y * w + prv.y * (1.f - w);
        sBint[g]      = make_float2(x, y);
        sBint[GG + g] = make_float2(x*x + y*y, 0.f);
    }
    __syncthreads();

    // =========== Phase 1: pred rows vs 1280 interp-gt cols ===========
    // score = |g|^2 - 2 p.g  (same argmin as squared distance)
    v2f A;
    {
        int row = wave * 16 + l;
        float2 p = sBpred[row];                  // unconditional load
        A.x = half ? 1.f : -2.f * p.x;           // cndmask, no exec branch
        A.y = half ? 0.f : -2.f * p.y;
    }
    float bestV[8]; int bestI[8];
#pragma unroll
    for (int r = 0; r < 8; ++r) { bestV[r] = 3.0e38f; bestI[r] = 0; }

    const int bOffI = half * GG + l;             // loop-invariant addr part
    {
        float2 bv = sBint[bOffI];                // prologue fetch (tile 0)
        for (int ct = 0; ct < GG / 16; ++ct) {
            v2f Bv; Bv.x = bv.x; Bv.y = bv.y;
            // prefetch next tile's B operand before consuming the WMMA
            float2 bvNext = sBint[bOffI + (((ct + 1) < GG / 16) ? (ct + 1) : ct) * 16];
            v8f C = {};
            C = __builtin_amdgcn_wmma_f32_16x16x4_f32(false, A, false, Bv,
                                                      (short)0, C, false, false);
            int idx = ct * 16 + l;
#pragma unroll
            for (int r = 0; r < 8; ++r) {
                float v = C[r];
                // idx strictly increases per lane => strict '<' keeps 1st min
                bool better = v < bestV[r];
                bestV[r] = better ? v : bestV[r];
                bestI[r] = better ? idx : bestI[r];
            }
            bv = bvNext;
        }
    }
#pragma unroll
    for (int r = 0; r < 8; ++r) {   // reduce across 16 lanes of each half
        float v = bestV[r]; int i = bestI[r];
#pragma unroll
        for (int ms = 1; ms <= 8; ms <<= 1) {
            float ov = __shfl_xor(v, ms, 32);
            int   oi = __shfl_xor(i, ms, 32);
            argminStep(v, i, ov, oi);
        }
        bestV[r] = v; bestI[r] = i;
    }
    if (l < 8) {  // lane l emits reg r=l; rows M = half*8 + r within tile
        int row = wave * 16 + half * 8 + l;
        float2 gp = sBint[bestI[l]];
        float2 pp = sPredPoly[row];
        c1[row] = fabsf(pp.x - gp.x) + fabsf(pp.y - gp.y);
    }

    // =========== Phase 2: gt rows vs 128 pred cols ===========
    v2f A2;
    {
        int row = wave * 16 + l;
        float2 g = sGt[row];
        A2.x = half ? 1.f : -2.f * g.x;
        A2.y = half ? 0.f : -2.f * g.y;
    }
#pragma unroll
    for (int r = 0; r < 8; ++r) { bestV[r] = 3.0e38f; bestI[r] = 0; }

    const int bOffP = half * NN + l;
    {
        float2 bv = sBpred[bOffP];
        for (int ct = 0; ct < NN / 16; ++ct) {
            v2f Bv; Bv.x = bv.x; Bv.y = bv.y;
            float2 bvNext = sBpred[bOffP + (((ct + 1) < NN / 16) ? (ct + 1) : ct) * 16];
            v8f C = {};
            C = __builtin_amdgcn_wmma_f32_16x16x4_f32(false, A2, false, Bv,
                                                      (short)0, C, false, false);
            int idx = ct * 16 + l;
#pragma unroll
            for (int r = 0; r < 8; ++r) {
                float v = C[r];
                bool better = v < bestV[r];
                bestV[r] = better ? v : bestV[r];
                bestI[r] = better ? idx : bestI[r];
            }
            bv = bvNext;
        }
    }
#pragma unroll
    for (int r = 0; r < 8; ++r) {
        float v = bestV[r]; int i = bestI[r];
#pragma unroll
        for (int ms = 1; ms <= 8; ms <<= 1) {
            float ov = __shfl_xor(v, ms, 32);
            int   oi = __shfl_xor(i, ms, 32);
            argminStep(v, i, ov, oi);
        }
        bestV[r] = v; bestI[r] = i;
    }
    if (l < 8) {
        int m = wave * 16 + half * 8 + l;
        float2 np = sPredPoly[bestI[l]];
        float2 g  = sGt[m];
        float mk  = kpMask[b * MM + m];
        c2[m] = (fabsf(np.x - g.x) + fabsf(np.y - g.y)) * mk;
        c3[m] = 2.f * mk;   // mask broadcast to 2 channels
    }
    __syncthreads();

    // ---- deterministic in-place tree reductions (128 entries each) ----
    for (int s = 64; s > 0; s >>= 1) {
        if (tid < s) {
            c1[tid] += c1[tid + s];
            c2[tid] += c2[tid + s];
            c3[tid] += c3[tid + s];
        }
        __syncthreads();
    }
    if (tid == 0) {
        ws[b]          = c1[0];
        ws[BB + b]     = c2[0];
        ws[2 * BB + b] = c3[0];
    }
}

__global__ __launch_bounds__(256) void dmloss_final_kernel(
    const float* __restrict__ ws, float* __restrict__ out)
{
    __shared__ float r1[256], r2[256], r3[256];
    int tid = threadIdx.x;
    float s1 = 0.f, s2 = 0.f, s3 = 0.f;
    for (int i = tid; i < BB; i += 256) {
        s1 += ws[i];
        s2 += ws[BB + i];
        s3 += ws[2 * BB + i];
    }
    r1[tid] = s1; r2[tid] = s2; r3[tid] = s3;
    __syncthreads();
    for (int s = 128; s > 0; s >>= 1) {
        if (tid < s) {
            r1[tid] += r1[tid + s];
            r2[tid] += r2[tid + s];
            r3[tid] += r3[tid + s];
        }
        __syncthreads();
    }
    if (tid == 0) {
        float lossPred2Gt = r1[0] / (float)(BB * NN * 2); // mean over B*N*2
        float lossSet = r2[0] / (r3[0] + 1.f) + lossPred2Gt;
        out[0] = lossSet * 0.5f;
    }
}

extern "C" void kernel_launch(void* const* d_in, const int* in_sizes, int n_in,
                              void* d_out, int out_size, void* d_ws, size_t ws_size,
                              hipStream_t stream) {
    const float* iniPred  = (const float*)d_in[0];
    const float* predPoly = (const float*)d_in[1];
    const float* gtPoly   = (const float*)d_in[2];
    const float* kpMask   = (const float*)d_in[3];
    float* ws  = (float*)d_ws;   // needs 3*1024 floats = 12 KB
    float* out = (float*)d_out;

    dmloss_main_kernel<<<BB, 256, 0, stream>>>(iniPred, predPoly, gtPoly, kpMask, ws);
    dmloss_final_kernel<<<1, 256, 0, stream>>>(ws, out);
}